// HyperbolicGCN_90177133346928
// MI455X (gfx1250) — compile-verified
//
#include <hip/hip_runtime.h>

#define EPS 1e-5f
#define N_NODES 2048
#define FDEG 20
#define DH 64

typedef __attribute__((ext_vector_type(2))) float v2f;
typedef __attribute__((ext_vector_type(8))) float v8f;

__device__ __forceinline__ float clampf(float x, float lo, float hi) {
    return fminf(fmaxf(x, lo), hi);
}
__device__ __forceinline__ float artanh_c(float x) {
    return atanhf(clampf(x, -1.0f + EPS, 1.0f - EPS));
}

// ---------------------------------------------------------------------------
// Kernel 1: f = exp0(concat(emb0[x0], emb1[x1]) @ win + bin)   [N,64]x[64,64]
// 128 threads = 4 waves; each block does 16 rows, each wave a 16-col tile.
// fp32 WMMA 16x16x4, 16 K-steps.
// ---------------------------------------------------------------------------
__global__ void __launch_bounds__(128)
featurize_kernel(const int* __restrict__ x,
                 const float* __restrict__ emb0,
                 const float* __restrict__ emb1,
                 const float* __restrict__ win,
                 const float* __restrict__ bin,
                 float* __restrict__ f)
{
    __shared__ float As[16][DH + 1];
    __shared__ float Hs[16][DH + 1];
    __shared__ float rsq[16];
    const int tid  = threadIdx.x;
    const int tile = blockIdx.x;

    for (int p = tid; p < 16 * DH; p += 128) {
        int r = p >> 6, c = p & 63;
        int node = tile * 16 + r;
        float v;
        if (c < 32) v = emb0[x[2 * node] * 32 + c];
        else        v = emb1[x[2 * node + 1] * 32 + (c - 32)];
        As[r][c] = v;
    }
    __syncthreads();

    const int lane = tid & 31;
    const int kh   = lane >> 4;      // K-half select (lanes 16-31 hold K+2)
    const int ln   = lane & 15;      // A row / B column within tile
    const int colBase = (tid >> 5) * 16;

    v8f acc = {};
    #pragma unroll
    for (int k0 = 0; k0 < DH; k0 += 4) {
        v2f a, b;
        a.x = As[ln][k0 + 2 * kh];
        a.y = As[ln][k0 + 2 * kh + 1];
        b.x = win[(k0 + 2 * kh) * DH + colBase + ln];
        b.y = win[(k0 + 2 * kh + 1) * DH + colBase + ln];
        acc = __builtin_amdgcn_wmma_f32_16x16x4_f32(
            false, a, false, b, (short)0, acc, false, false);
    }
    #pragma unroll
    for (int r = 0; r < 8; ++r) {
        int row = r + 8 * kh;
        int col = colBase + ln;
        Hs[row][col] = acc[r] + bin[col];
    }
    __syncthreads();

    if (tid < 16) {
        float s = 0.f;
        for (int c = 0; c < DH; ++c) s += Hs[tid][c] * Hs[tid][c];
        rsq[tid] = s;
    }
    __syncthreads();
    for (int p = tid; p < 16 * DH; p += 128) {
        int r = p >> 6, c = p & 63;
        float nrm = sqrtf(rsq[r] + 1e-15f);
        float sc  = tanhf(nrm) / fmaxf(nrm, EPS);
        f[(tile * 16 + r) * DH + c] = sc * Hs[r][c];
    }
}

// ---------------------------------------------------------------------------
// Kernel 3: out = exp0(u @ wout + bout)   [N,64]x[64,64], same WMMA scheme.
// ---------------------------------------------------------------------------
__global__ void __launch_bounds__(128)
out_kernel(const float* __restrict__ u,
           const float* __restrict__ wout,
           const float* __restrict__ bout,
           float* __restrict__ out)
{
    __shared__ float As[16][DH + 1];
    __shared__ float Hs[16][DH + 1];
    __shared__ float rsq[16];
    const int tid  = threadIdx.x;
    const int tile = blockIdx.x;

    for (int p = tid; p < 16 * DH; p += 128) {
        int r = p >> 6, c = p & 63;
        As[r][c] = u[(tile * 16 + r) * DH + c];
    }
    __syncthreads();

    const int lane = tid & 31;
    const int kh   = lane >> 4;
    const int ln   = lane & 15;
    const int colBase = (tid >> 5) * 16;

    v8f acc = {};
    #pragma unroll
    for (int k0 = 0; k0 < DH; k0 += 4) {
        v2f a, b;
        a.x = As[ln][k0 + 2 * kh];
        a.y = As[ln][k0 + 2 * kh + 1];
        b.x = wout[(k0 + 2 * kh) * DH + colBase + ln];
        b.y = wout[(k0 + 2 * kh + 1) * DH + colBase + ln];
        acc = __builtin_amdgcn_wmma_f32_16x16x4_f32(
            false, a, false, b, (short)0, acc, false, false);
    }
    #pragma unroll
    for (int r = 0; r < 8; ++r) {
        int row = r + 8 * kh;
        int col = colBase + ln;
        Hs[row][col] = acc[r] + bout[col];
    }
    __syncthreads();

    if (tid < 16) {
        float s = 0.f;
        for (int c = 0; c < DH; ++c) s += Hs[tid][c] * Hs[tid][c];
        rsq[tid] = s;
    }
    __syncthreads();
    for (int p = tid; p < 16 * DH; p += 128) {
        int r = p >> 6, c = p & 63;
        float nrm = sqrtf(rsq[r] + 1e-15f);
        float sc  = tanhf(nrm) / fmaxf(nrm, EPS);
        out[(tile * 16 + r) * DH + c] = sc * Hs[r][c];
    }
}

// ---------------------------------------------------------------------------
// _aggregate(h_self=fv, h_neigh=H[20][65], scalar r) -> zout = log0(e)
// ---------------------------------------------------------------------------
__device__ void aggregate_block(const float* __restrict__ H, // [FDEG*(DH+1)] LDS
                                const float* fv, float fsq, float r_scalar,
                                float* hsqS, float* rowS, float* rowS2,
                                float* vecS, float* scalS, float* zout, int tid)
{
    const int LD = DH + 1;
    if (tid < FDEG) {
        float s = 0.f;
        for (int c = 0; c < DH; ++c) { float v = H[tid * LD + c]; s += v * v; }
        hsqS[tid] = s;
    }
    __syncthreads();
    if (tid < FDEG) {
        float d2 = 0.f;
        for (int c = 0; c < DH; ++c) { float d = fv[c] - H[tid * LD + c]; d2 += d * d; }
        float den = fmaxf((1.f - fsq) * (1.f - hsqS[tid]), EPS);
        rowS[tid] = acoshf(fmaxf(1.f + 2.f * d2 / den, 1.f + 1e-7f));
    }
    __syncthreads();
    if (tid == 0) {
        float mx = -1e30f;
        for (int i = 0; i < FDEG; ++i) mx = fmaxf(mx, -rowS[i]);
        float sum = 0.f;
        for (int i = 0; i < FDEG; ++i) { float e = expf(-rowS[i] - mx); rowS2[i] = e; sum += e; }
        for (int i = 0; i < FDEG; ++i) {
            float att = rowS2[i] / sum;
            float nh  = clampf(sqrtf(hsqS[i] + 1e-15f), EPS, 1.f - EPS);
            float t   = tanhf(att * artanh_c(nh));
            float ratio = t / nh;
            float nm  = clampf(sqrtf(ratio * ratio * hsqS[i] + 1e-15f), EPS, 1.f - EPS);
            rowS[i]   = artanh_c(nm) / nm * ratio;   // scalar coeff on H_i
        }
    }
    __syncthreads();
    {   // nTE (pre-exp0): vecS[c] = sum_i alpha_i * H[i][c]
        float s = 0.f;
        for (int i = 0; i < FDEG; ++i) s += rowS[i] * H[i * LD + tid];
        vecS[tid] = s;
    }
    __syncthreads();
    if (tid == 0) {
        float q = 0.f;
        for (int c = 0; c < DH; ++c) q += vecS[c] * vecS[c];
        float nrm = sqrtf(q + 1e-15f);
        float sc  = tanhf(nrm) / fmaxf(nrm, EPS);     // exp0 scale
        float y2  = sc * sc * q;
        float nf  = clampf(sqrtf(fsq + 1e-15f), EPS, 1.f - EPS);
        float cF  = tanhf(r_scalar * artanh_c(nf)) / nf;
        float dot = 0.f;
        for (int c = 0; c < DH; ++c) dot += fv[c] * vecS[c];
        float xy  = cF * dot * sc;
        float x2  = cF * cF * fsq;
        float den = fmaxf(1.f + 2.f * xy + x2 * y2, EPS);
        scalS[5] = (1.f + 2.f * xy + y2) * cF / den;  // coeff on fv
        scalS[6] = (1.f - x2) * sc / den;             // coeff on raw vecS
    }
    __syncthreads();
    vecS[tid] = scalS[5] * fv[tid] + scalS[6] * vecS[tid];   // e
    __syncthreads();
    if (tid == 0) {
        float q = 0.f;
        for (int c = 0; c < DH; ++c) q += vecS[c] * vecS[c];
        float nn = clampf(sqrtf(q + 1e-15f), EPS, 1.f - EPS);
        scalS[7] = artanh_c(nn) / nn;
    }
    __syncthreads();
    zout[tid] = scalS[7] * vecS[tid];
    __syncthreads();
}

// ---------------------------------------------------------------------------
// Kernel 2: per-node coattention + aggregation + beta-MLP; writes u=log0(rst).
// One 64-thread block per node; all intermediates in LDS.
// ---------------------------------------------------------------------------
__global__ void __launch_bounds__(64)
coatt_kernel(const float* __restrict__ f,
             const int* __restrict__ Didx,
             const int* __restrict__ Qidx,
             const float* __restrict__ qp,
             const float* __restrict__ qqp,
             const float* __restrict__ q1p,
             const float* __restrict__ aw1,
             const float* __restrict__ ab1,
             const float* __restrict__ aw2,
             float* __restrict__ u)
{
    const int LD = DH + 1;
    __shared__ float fv[DH];
    __shared__ float Dv[FDEG][DH + 1];
    __shared__ float Qv[FDEG][DH + 1];
    __shared__ float Sv[FDEG][DH + 1];
    __shared__ float pairS[FDEG * FDEG];
    __shared__ float dsq[FDEG], qsq[FDEG];
    __shared__ float rowA[FDEG], rowB[FDEG], hsqS[FDEG];
    __shared__ float vecS[DH], z0[DH], z1[DH];
    __shared__ float scalS[8];

    const int n = blockIdx.x;
    const int tid = threadIdx.x;

    fv[tid] = f[n * DH + tid];
    for (int p = tid; p < FDEG * DH; p += 64) {
        int i = p >> 6, c = p & 63;
        Dv[i][c] = f[Didx[n * FDEG + i] * DH + c];
        Qv[i][c] = f[Qidx[n * FDEG + i] * DH + c];
    }
    __syncthreads();

    if (tid < FDEG) {
        float s = 0.f;
        for (int c = 0; c < DH; ++c) s += Dv[tid][c] * Dv[tid][c];
        dsq[tid] = s;
    } else if (tid >= 32 && tid < 32 + FDEG) {
        int j = tid - 32;
        float s = 0.f;
        for (int c = 0; c < DH; ++c) s += Qv[j][c] * Qv[j][c];
        qsq[j] = s;
    } else if (tid == 63) {
        float s = 0.f;
        for (int c = 0; c < DH; ++c) s += fv[c] * fv[c];
        scalS[0] = s;
    }
    __syncthreads();

    // pairwise Poincare distances D_i vs Q_j
    for (int p = tid; p < FDEG * FDEG; p += 64) {
        int i = p / FDEG, j = p % FDEG;
        float d2 = 0.f;
        for (int c = 0; c < DH; ++c) { float d = Dv[i][c] - Qv[j][c]; d2 += d * d; }
        float den = fmaxf((1.f - dsq[i]) * (1.f - qsq[j]), EPS);
        pairS[p] = acoshf(fmaxf(1.f + 2.f * d2 / den, 1.f + 1e-7f));
    }
    __syncthreads();

    // row softmax over Q neighbors, fold into scalar alpha_ij
    if (tid < FDEG) {
        int i = tid;
        float e[FDEG];
        float mx = -1e30f;
        for (int j = 0; j < FDEG; ++j) mx = fmaxf(mx, -pairS[i * FDEG + j]);
        float sum = 0.f;
        for (int j = 0; j < FDEG; ++j) { e[j] = expf(-pairS[i * FDEG + j] - mx); sum += e[j]; }
        for (int j = 0; j < FDEG; ++j) {
            float G  = e[j] / sum;
            float nq = clampf(sqrtf(qsq[j] + 1e-15f), EPS, 1.f - EPS);
            float t  = tanhf(G * artanh_c(nq));
            float ratio = t / nq;
            float nm = clampf(sqrtf(ratio * ratio * qsq[j] + 1e-15f), EPS, 1.f - EPS);
            pairS[i * FDEG + j] = artanh_c(nm) / nm * ratio;
        }
    }
    __syncthreads();

    // S = alpha @ Q  (pre-exp0 tangent sum)
    for (int p = tid; p < FDEG * DH; p += 64) {
        int i = p >> 6, c = p & 63;
        float s = 0.f;
        for (int j = 0; j < FDEG; ++j) s += pairS[i * FDEG + j] * Qv[j][c];
        Sv[i][c] = s;
    }
    __syncthreads();
    if (tid < FDEG) {
        float s = 0.f;
        for (int c = 0; c < DH; ++c) s += Sv[tid][c] * Sv[tid][c];
        float nrm = sqrtf(s + 1e-15f);
        float sc  = tanhf(nrm) / fmaxf(nrm, EPS);
        rowA[tid] = sc;            // nh scale
        rowB[tid] = sc * sc * s;   // y2 = |nh|^2
    }
    __syncthreads();
    for (int p = tid; p < FDEG * DH; p += 64) {
        int i = p >> 6, c = p & 63;
        Sv[i][c] *= rowA[i];       // Sv now holds nh
    }
    __syncthreads();

    // h_neigh_i = mob_add(mob_scalar(D_i, q1), nh_i)  -> overwrite Dv
    const float q1 = q1p[0];
    if (tid < FDEG) {
        int i = tid;
        float nD = clampf(sqrtf(dsq[i] + 1e-15f), EPS, 1.f - EPS);
        float cD = tanhf(q1 * artanh_c(nD)) / nD;
        float dot = 0.f;
        for (int c = 0; c < DH; ++c) dot += Dv[i][c] * Sv[i][c];
        float xy = cD * dot;
        float x2 = cD * cD * dsq[i];
        float y2 = rowB[i];
        float den = fmaxf(1.f + 2.f * xy + x2 * y2, EPS);
        rowA[i] = (1.f + 2.f * xy + y2) * cD / den;
        rowB[i] = (1.f - x2) / den;
    }
    __syncthreads();
    for (int p = tid; p < FDEG * DH; p += 64) {
        int i = p >> 6, c = p & 63;
        Dv[i][c] = rowA[i] * Dv[i][c] + rowB[i] * Sv[i][c];
    }
    __syncthreads();

    const float fsq = scalS[0];
    aggregate_block(&Dv[0][0], fv, fsq, qp[0],  hsqS, rowA, rowB, vecS, scalS, z0, tid);
    aggregate_block(&Qv[0][0], fv, fsq, qqp[0], hsqS, rowA, rowB, vecS, scalS, z1, tid);

    // beta-MLP: tanh(z @ aw1 + ab1) @ aw2, softmax over the 2 rows
    {
        int rrow = tid >> 5;
        int t = tid & 31;
        const float* zr = (rrow == 0) ? z0 : z1;
        float s = ab1[t];
        for (int c = 0; c < DH; ++c) s += zr[c] * aw1[c * 32 + t];
        vecS[tid] = tanhf(s);
    }
    __syncthreads();
    if (tid == 0) {
        float l0 = 0.f, l1 = 0.f;
        for (int k = 0; k < 32; ++k) { l0 += vecS[k] * aw2[k]; l1 += vecS[32 + k] * aw2[k]; }
        float mx = fmaxf(l0, l1);
        float e0 = expf(l0 - mx), e1v = expf(l1 - mx);
        float inv = 1.f / (e0 + e1v);
        scalS[0] = e0 * inv;
        scalS[1] = e1v * inv;
    }
    __syncthreads();
    vecS[tid] = scalS[0] * z0[tid] + scalS[1] * z1[tid];
    __syncthreads();
    if (tid == 0) {
        float q = 0.f;
        for (int c = 0; c < DH; ++c) q += vecS[c] * vecS[c];
        float nrm = sqrtf(q + 1e-15f);
        float sc  = tanhf(nrm) / fmaxf(nrm, EPS);        // exp0 -> rst
        float q2  = sc * sc * q;
        float n2  = clampf(sqrtf(q2 + 1e-15f), EPS, 1.f - EPS);
        scalS[2]  = artanh_c(n2) / n2 * sc;              // log0(exp0(.)) scale
    }
    __syncthreads();
    u[n * DH + tid] = scalS[2] * vecS[tid];
    (void)LD;
}

// ---------------------------------------------------------------------------
extern "C" void kernel_launch(void* const* d_in, const int* in_sizes, int n_in,
                              void* d_out, int out_size, void* d_ws, size_t ws_size,
                              hipStream_t stream)
{
    (void)in_sizes; (void)n_in; (void)out_size; (void)ws_size;

    const int* x       = (const int*)d_in[0];
    const int* idx_sim = (const int*)d_in[1];
    const int* idx_cor = (const int*)d_in[2];

    float* f_ws = (float*)d_ws;                  // N*DH floats
    float* u_ws = f_ws + N_NODES * DH;           // N*DH floats
    float* out  = (float*)d_out;

    for (int mode = 0; mode < 2; ++mode) {
        const int base = 3 + mode * 12;
        const float* emb0 = (const float*)d_in[base + 0];
        const float* emb1 = (const float*)d_in[base + 1];
        const float* win  = (const float*)d_in[base + 2];
        const float* bin  = (const float*)d_in[base + 3];
        const float* q    = (const float*)d_in[base + 4];
        const float* qq   = (const float*)d_in[base + 5];
        const float* q1   = (const float*)d_in[base + 6];
        const float* aw1  = (const float*)d_in[base + 7];
        const float* ab1  = (const float*)d_in[base + 8];
        const float* aw2  = (const float*)d_in[base + 9];
        const float* wout = (const float*)d_in[base + 10];
        const float* bout = (const float*)d_in[base + 11];
        // mode 'sim': D = f[idx_cor], Q = f[idx_sim]; mode 'cor': swapped
        const int* Didx = (mode == 0) ? idx_cor : idx_sim;
        const int* Qidx = (mode == 0) ? idx_sim : idx_cor;

        featurize_kernel<<<N_NODES / 16, 128, 0, stream>>>(x, emb0, emb1, win, bin, f_ws);
        coatt_kernel<<<N_NODES, 64, 0, stream>>>(f_ws, Didx, Qidx, q, qq, q1,
                                                 aw1, ab1, aw2, u_ws);
        out_kernel<<<N_NODES / 16, 128, 0, stream>>>(u_ws, wout, bout,
                                                     out + (size_t)mode * N_NODES * DH);
    }
}